// ModulatedDeformConvPack_62466004353750
// MI455X (gfx1250) — compile-verified
//
#include <hip/hip_runtime.h>
#include <hip/hip_bf16.h>

// MI455X (gfx1250) modulated deformable conv.
// Heavy GEMM (M=64, K=576, N=73728) on V_WMMA_F32_16X16X4_F32 (wave32 fp32
// matrix path, full fp32 fidelity). Two pixel-tiles per block so each A
// (weight) fragment loaded from L2 feeds two WMMAs -> 2x arithmetic
// intensity on the global path and two independent accumulator chains.

#define B_   2
#define CIN  64
#define COUT 64
#define H_   192
#define W_   192
#define KK   9
#define NC_  20
#define HW   (H_ * W_)
#define KDIM (CIN * KK)        // 576
#define SB_STRIDE 578          // 576 + 2 pad: even (8B-aligned), bank spread
#define TILE_PIX 32            // two 16-pixel WMMA tiles per block

typedef __attribute__((ext_vector_type(2))) float v2f;
typedef __attribute__((ext_vector_type(8))) float v8f;

// ---------------------------------------------------------------- kernel 1
// offset = conv3x3(x, off_w) + off_b    [B,18,H,W]
__global__ void dcn_offset_conv(const float* __restrict__ x,
                                const float* __restrict__ off_w,
                                const float* __restrict__ off_b,
                                float* __restrict__ off_out) {
  int idx = blockIdx.x * blockDim.x + threadIdx.x;
  const int total = B_ * 18 * HW;
  if (idx >= total) return;
  int s  = idx % HW;
  int oc = (idx / HW) % 18;
  int b  = idx / (HW * 18);
  int h = s / W_, w = s % W_;
  float acc = off_b[oc];
  const float* xb = x + (size_t)b * CIN * HW;
  const float* wb = off_w + (size_t)oc * CIN * 9;
#pragma unroll
  for (int kh = 0; kh < 3; ++kh) {
    int ih = h - 1 + kh;
    if (ih < 0 || ih >= H_) continue;
#pragma unroll
    for (int kw = 0; kw < 3; ++kw) {
      int iw = w - 1 + kw;
      if (iw < 0 || iw >= W_) continue;
      int sp = ih * W_ + iw;
      int wo = kh * 3 + kw;
      for (int c = 0; c < CIN; ++c)
        acc = fmaf(xb[(size_t)c * HW + sp], wb[c * 9 + wo], acc);
    }
  }
  off_out[idx] = acc;
}

// ---------------------------------------------------------------- kernel 2
// mask[b,k,h,w] = sum_n bilinear_clamp(input_LO)[b,n] * output_LO[b,n,h,w]
__global__ void dcn_mask(const float* __restrict__ in_LO,
                         const float* __restrict__ out_LO,
                         const float* __restrict__ off,
                         float* __restrict__ mask) {
  int idx = blockIdx.x * blockDim.x + threadIdx.x;
  const int total = B_ * KK * HW;
  if (idx >= total) return;
  int s = idx % HW;
  int k = (idx / HW) % KK;
  int b = idx / (HW * KK);
  int h = s / W_, w = s % W_;
  float oy = off[((size_t)(b * 18 + 2 * k)) * HW + s];
  float ox = off[((size_t)(b * 18 + 2 * k + 1)) * HW + s];
  float px = (float)w - 0.5f * (float)(W_ - 1) + (float)(k % 3) + ox;
  float py = (float)h - 0.5f * (float)(H_ - 1) + (float)(k / 3) + oy;
  px = fminf(fmaxf(px, 0.0f), (float)(W_ - 1));   // border clamp
  py = fminf(fmaxf(py, 0.0f), (float)(H_ - 1));
  float x0f = floorf(px), y0f = floorf(py);
  float lx = px - x0f, ly = py - y0f;
  int x0 = (int)x0f, y0 = (int)y0f;
  int x1 = min(x0 + 1, W_ - 1), y1 = min(y0 + 1, H_ - 1);
  int i00 = y0 * W_ + x0, i01 = y0 * W_ + x1;
  int i10 = y1 * W_ + x0, i11 = y1 * W_ + x1;
  float w00 = (1.f - ly) * (1.f - lx), w01 = (1.f - ly) * lx;
  float w10 = ly * (1.f - lx),        w11 = ly * lx;
  const float* ib = in_LO  + (size_t)b * NC_ * HW;
  const float* ob = out_LO + (size_t)b * NC_ * HW;
  float acc = 0.f;
  for (int n = 0; n < NC_; ++n) {
    const float* p = ib + (size_t)n * HW;
    float sv = p[i00] * w00 + p[i01] * w01 + p[i10] * w10 + p[i11] * w11;
    acc = fmaf(sv, ob[(size_t)n * HW + s], acc);
  }
  mask[idx] = acc;
}

// ---------------------------------------------------------------- kernel 3
// Fused deformable sample + modulation + GEMM on WMMA.
// One block = 32 output pixels (two 16-wide WMMA N-tiles). 128 threads =
// 4 wave32; wave wv owns output channels [16*wv, 16*wv+16) and computes
// BOTH pixel tiles from one A-fragment load (2x intensity, 2 acc chains).
__global__ __launch_bounds__(128) void dcn_main_wmma(
    const float* __restrict__ x,
    const float* __restrict__ weight,   // [COUT][CIN][9]
    const float* __restrict__ bias,     // [COUT]
    const float* __restrict__ off,      // [B][18][HW]
    const float* __restrict__ mask,     // [B][9][HW]
    float* __restrict__ out)            // [B][COUT][HW]
{
  __shared__ int   sIdx[TILE_PIX * KK * 4];     // 288*4 ints
  __shared__ float sWt [TILE_PIX * KK * 4];
  __shared__ float sB  [TILE_PIX * SB_STRIDE];  // [pixel][row r = c*9+k] ~74KB

  const int t    = threadIdx.x;
  const int tile = blockIdx.x;
  const int b    = tile / (HW / TILE_PIX);
  const int s0   = (tile % (HW / TILE_PIX)) * TILE_PIX;

  // --- phase A: per (pixel, k) bilinear corner indices & (weight * mask)
  for (int pp = t; pp < TILE_PIX * KK; pp += 128) {
    int p = pp / KK, k = pp % KK;
    int s = s0 + p;
    int h = s / W_, w = s % W_;
    float oy = off[((size_t)(b * 18 + 2 * k)) * HW + s];
    float ox = off[((size_t)(b * 18 + 2 * k + 1)) * HW + s];
    float m  = mask[((size_t)(b * KK + k)) * HW + s];
    float py = (float)(h - 1 + k / 3) + oy;       // zero-pad sampling coords
    float px = (float)(w - 1 + k % 3) + ox;
    float y0f = floorf(py), x0f = floorf(px);
    float ly = py - y0f, lx = px - x0f;
    int y0 = (int)y0f, x0 = (int)x0f;
    int y1 = y0 + 1,   x1 = x0 + 1;
    float vy0 = (y0 >= 0 && y0 < H_) ? 1.f : 0.f;
    float vy1 = (y1 >= 0 && y1 < H_) ? 1.f : 0.f;
    float vx0 = (x0 >= 0 && x0 < W_) ? 1.f : 0.f;
    float vx1 = (x1 >= 0 && x1 < W_) ? 1.f : 0.f;
    int cy0 = min(max(y0, 0), H_ - 1), cy1 = min(max(y1, 0), H_ - 1);
    int cx0 = min(max(x0, 0), W_ - 1), cx1 = min(max(x1, 0), W_ - 1);
    sIdx[pp * 4 + 0] = cy0 * W_ + cx0;
    sIdx[pp * 4 + 1] = cy0 * W_ + cx1;
    sIdx[pp * 4 + 2] = cy1 * W_ + cx0;
    sIdx[pp * 4 + 3] = cy1 * W_ + cx1;
    sWt[pp * 4 + 0] = (1.f - ly) * (1.f - lx) * vy0 * vx0 * m;
    sWt[pp * 4 + 1] = (1.f - ly) * lx         * vy0 * vx1 * m;
    sWt[pp * 4 + 2] = ly * (1.f - lx)         * vy1 * vx0 * m;
    sWt[pp * 4 + 3] = ly * lx                 * vy1 * vx1 * m;
  }
  __syncthreads();

  // --- phase B: gather-modulate B-matrices into LDS (x is L2-resident)
  {
    const int ch0 = t & 7;        // 8 channel lanes per (p,k)
    const int g   = t >> 3;       // 16 (p,k) groups per pass
    const float* xb = x + (size_t)b * CIN * HW;
    for (int j = 0; j < 2 * KK; ++j) {
      int pk = g + 16 * j;        // covers all 288 (p,k)
      int p = pk / KK, k = pk % KK;
      int i0 = sIdx[pk * 4 + 0], i1 = sIdx[pk * 4 + 1];
      int i2 = sIdx[pk * 4 + 2], i3 = sIdx[pk * 4 + 3];
      float w0 = sWt[pk * 4 + 0], w1 = sWt[pk * 4 + 1];
      float w2 = sWt[pk * 4 + 2], w3 = sWt[pk * 4 + 3];
#pragma unroll
      for (int cc = 0; cc < 8; ++cc) {
        int c = ch0 + 8 * cc;
        const float* pl = xb + (size_t)c * HW;
        float v = pl[i0] * w0 + pl[i1] * w1 + pl[i2] * w2 + pl[i3] * w3;
        sB[p * SB_STRIDE + c * KK + k] = v;
      }
    }
  }
  __syncthreads();

  // --- phase C: WMMA f32 16x16x4, one A-fragment -> two B tiles
  // A layout (ISA 7.12.2, 32-bit 16x4): lane holds m=lane%16; elem j -> K=2*(lane/16)+j
  // B layout: lane holds n=lane%16; elem j -> K=2*(lane/16)+j
  const int lane = t & 31;
  const int wv   = t >> 5;                 // 0..3 -> COUT slice
  const int mrow = lane & 15;
  const int koff = (lane >> 4) << 1;       // 0 or 2
  const float* wrow = weight + (size_t)(wv * 16 + mrow) * KDIM;
  const float* sB0  = &sB[mrow * SB_STRIDE];
  const float* sB1  = &sB[(mrow + 16) * SB_STRIDE];
  v8f acc0 = {}, acc1 = {};
#pragma unroll 4
  for (int r0 = 0; r0 < KDIM; r0 += 4) {
    v2f a, b0, b1;
    a.x  = wrow[r0 + koff];
    a.y  = wrow[r0 + koff + 1];
    b0.x = sB0[r0 + koff];
    b0.y = sB0[r0 + koff + 1];
    b1.x = sB1[r0 + koff];
    b1.y = sB1[r0 + koff + 1];
    acc0 = __builtin_amdgcn_wmma_f32_16x16x4_f32(
        false, a, false, b0, (short)0, acc0, false, false);
    acc1 = __builtin_amdgcn_wmma_f32_16x16x4_f32(
        false, a, false, b1, (short)0, acc1, false, false);
  }

  // --- phase D: C/D layout: n = lane%16 (pixel), m = v + 8*(lane/16)
  const int half = lane >> 4;
#pragma unroll
  for (int v = 0; v < 8; ++v) {
    int oc = wv * 16 + v + 8 * half;
    float bv = bias[oc];
    size_t base = ((size_t)(b * COUT + oc)) * HW + s0 + mrow;
    out[base]      = acc0[v] + bv;
    out[base + 16] = acc1[v] + bv;
  }
}

// ---------------------------------------------------------------- launch
extern "C" void kernel_launch(void* const* d_in, const int* in_sizes, int n_in,
                              void* d_out, int out_size, void* d_ws, size_t ws_size,
                              hipStream_t stream) {
  const float* x      = (const float*)d_in[0];
  const float* in_LO  = (const float*)d_in[1];
  const float* out_LO = (const float*)d_in[2];
  const float* weight = (const float*)d_in[3];
  const float* bias   = (const float*)d_in[4];
  const float* off_w  = (const float*)d_in[5];
  const float* off_b  = (const float*)d_in[6];
  float* out = (float*)d_out;

  // workspace: offsets [B,18,HW] then mask [B,9,HW]  (~8 MB total)
  float* off  = (float*)d_ws;
  float* mask = off + (size_t)B_ * 18 * HW;

  int n1 = B_ * 18 * HW;
  dcn_offset_conv<<<(n1 + 255) / 256, 256, 0, stream>>>(x, off_w, off_b, off);

  int n2 = B_ * KK * HW;
  dcn_mask<<<(n2 + 255) / 256, 256, 0, stream>>>(in_LO, out_LO, off, mask);

  dcn_main_wmma<<<B_ * (HW / TILE_PIX), 128, 0, stream>>>(x, weight, bias, off, mask, out);
}